// MultiBoxLoss_84971632984126
// MI455X (gfx1250) — compile-verified
//
#include <hip/hip_runtime.h>
#include <cstdint>

#define B_ 128
#define A_ 8732
#define C_ 21
#define NEG_RATIO_ 3
#define NTH 256
#define NBLK1 ((B_ * A_) / NTH)      // 4366 exactly (128*8732 = 1117696 = 4366*256)
#define PMWORDS ((A_ + 31) / 32)     // 273

// ---------- helpers ----------
__device__ __forceinline__ float smooth_l1(float d) {
    float ad = fabsf(d);
    return (ad < 1.0f) ? (0.5f * d * d) : (ad - 0.5f);
}

// deterministic fixed-order tree reduction (256 threads)
__device__ __forceinline__ float blk_reduce_f(float v, float* red, int tid) {
    red[tid] = v; __syncthreads();
    #pragma unroll
    for (int s = NTH / 2; s > 0; s >>= 1) {
        if (tid < s) red[tid] += red[tid + s];
        __syncthreads();
    }
    float r = red[0]; __syncthreads();
    return r;
}
__device__ __forceinline__ int blk_reduce_i(int v, int* red, int tid) {
    red[tid] = v; __syncthreads();
    #pragma unroll
    for (int s = NTH / 2; s > 0; s >>= 1) {
        if (tid < s) red[tid] += red[tid + s];
        __syncthreads();
    }
    int r = red[0]; __syncthreads();
    return r;
}

__device__ __forceinline__ float focal_from_ce(float ce) {
    float pt = expf(-ce);
    float om = 1.0f - pt;
    return om * om * ce;   // ALPHA=1, GAMMA=2
}

// CDNA5 async global->LDS copy (ASYNCcnt path)
__device__ __forceinline__ void async_ld_b128(unsigned lds_off, const void* g) {
    asm volatile("global_load_async_to_lds_b128 %0, %1, off"
                 :: "v"(lds_off), "v"((unsigned long long)(uintptr_t)g)
                 : "memory");
}
__device__ __forceinline__ void wait_async0() {
    asm volatile("s_wait_asynccnt 0" ::: "memory");
}

// ---------- K1: per-anchor CE + smooth-L1 partials ----------
// Each block async-stages its contiguous 256*21-float logits slab into LDS
// (b128/lane, fully coalesced), overlapping with label/loc loads, then does
// the 21-class logsumexp out of LDS (stride 21 => bank-conflict-free wave32).
__global__ __launch_bounds__(NTH) void k1_ce_loc(
    const float* __restrict__ pred_loc, const float* __restrict__ pred_label,
    const float* __restrict__ gt_loc, const int* __restrict__ gt_label,
    float* __restrict__ ce_out, float* __restrict__ loc_part) {
    __shared__ __align__(16) float slog[NTH * C_];   // 21504 B
    __shared__ float red_f[NTH];
    const int tid = threadIdx.x;
    const int idx0 = blockIdx.x * NTH;
    const int idx = idx0 + tid;          // grid exact: idx < B_*A_ always

    // issue async bulk copy of this block's logits first (ASYNCcnt)
    {
        const char* g = (const char*)(pred_label + (long)idx0 * C_);
        unsigned base = (unsigned)(uintptr_t)&slog[0];
        for (int i = tid; i < (NTH * C_ * 4) / 16; i += NTH)   // 1344 chunks
            async_ld_b128(base + 16u * i, g + 16 * i);
    }

    // overlap: labels + smooth-L1 while the DMA is in flight
    int lbl = gt_label[idx];
    float locp = 0.0f;
    if (lbl > 0) {
        float4 p = ((const float4*)pred_loc)[idx];
        float4 g = ((const float4*)gt_loc)[idx];
        locp = smooth_l1(p.x - g.x) + smooth_l1(p.y - g.y) +
               smooth_l1(p.z - g.z) + smooth_l1(p.w - g.w);
    }

    wait_async0();
    __syncthreads();

    const float* lg = &slog[tid * C_];
    float xs[C_];
    float m = -3.402823466e38f, xl = 0.0f;
    #pragma unroll
    for (int c = 0; c < C_; ++c) {
        float x = lg[c];
        xs[c] = x;
        m = fmaxf(m, x);
        if (c == lbl) xl = x;            // avoids dynamic register indexing
    }
    float s = 0.0f;
    #pragma unroll
    for (int c = 0; c < C_; ++c) s += expf(xs[c] - m);
    float ce = m + logf(s) - xl;         // logsumexp - logit[label]
    ce_out[idx] = ce;

    float tot = blk_reduce_f(locp, red_f, tid);
    if (tid == 0) loc_part[blockIdx.x] = tot;
}

// ---------- K2: per-row hard-negative mining via LDS radix-select ----------
__global__ __launch_bounds__(NTH) void k2_mine(
    const float* __restrict__ ce_g, const int* __restrict__ gt_label,
    float* __restrict__ cls_out, int* __restrict__ np_out) {
    __shared__ __align__(16) float ces[A_];     // 34928 B
    __shared__ unsigned posmask[PMWORDS];       // 1092 B
    __shared__ unsigned hist[256];
    __shared__ float red_f[NTH];
    __shared__ int red_i[NTH];
    __shared__ unsigned s_b0, s_cet;
    __shared__ int s_rem;

    const int tid = threadIdx.x;
    const int row = blockIdx.x;
    const float* ce_row = ce_g + (long)row * A_;

    // Async-stage the ce row into LDS: 8732 floats = 2183 x b128, exact.
    {
        unsigned base = (unsigned)(uintptr_t)&ces[0];
        const char* g = (const char*)ce_row;
        for (int i = tid; i < (A_ * 4) / 16; i += NTH)
            async_ld_b128(base + 16u * i, g + 16 * i);
    }
    for (int i = tid; i < PMWORDS; i += NTH) posmask[i] = 0u;
    wait_async0();
    __syncthreads();

    const int* gl = gt_label + (long)row * A_;
    for (int i = tid; i < A_; i += NTH)
        if (gl[i] > 0) atomicOr(&posmask[i >> 5], 1u << (i & 31));
    __syncthreads();

    int npl = 0;
    for (int i = tid; i < PMWORDS; i += NTH) npl += __popc(posmask[i]);
    const int np = blk_reduce_i(npl, red_i, tid);
    const int k = min(NEG_RATIO_ * np, A_ - 1);

    // 4-pass radix select: k-th largest floss bit pattern (floss>=0 => uint order).
    unsigned prefix = 0u;
    if (k > 0) {
        int remaining = k;
        for (int pass = 0; pass < 4; ++pass) {
            const int shift = 24 - 8 * pass;
            for (int i = tid; i < 256; i += NTH) hist[i] = 0u;
            __syncthreads();
            const unsigned pm = (pass == 0) ? 0u : (0xFFFFFFFFu << (shift + 8));
            for (int i = tid; i < A_; i += NTH) {
                bool pos = (posmask[i >> 5] >> (i & 31)) & 1u;
                float fl = pos ? 0.0f : focal_from_ce(ces[i]);
                unsigned v = __float_as_uint(fl);
                if ((v & pm) == prefix) atomicAdd(&hist[(v >> shift) & 255u], 1u);
            }
            __syncthreads();
            if (tid == 0) {
                int cum = 0, b = 255;
                for (; b >= 0; --b) { cum += (int)hist[b]; if (cum >= remaining) break; }
                s_b0 = (unsigned)b;
                s_rem = remaining - (cum - (int)hist[b]);
            }
            __syncthreads();
            prefix |= (s_b0 << shift);
            remaining = s_rem;
            __syncthreads();
        }
    }
    const unsigned tbits = prefix;
    if (tid == 0) s_cet = 0u;
    __syncthreads();

    // Final pass: sum CE over pos and over strict top-k (> t), handle ties exactly.
    float sum_pos = 0.0f, sum_gt = 0.0f;
    int cgt = 0;
    for (int i = tid; i < A_; i += NTH) {
        bool pos = (posmask[i >> 5] >> (i & 31)) & 1u;
        float ce = ces[i];
        float fl = pos ? 0.0f : focal_from_ce(ce);
        unsigned v = __float_as_uint(fl);
        if (pos) sum_pos += ce;
        if (k > 0) {
            if (v > tbits) { sum_gt += ce; ++cgt; }   // pos has v==0, never > t
            else if (v == tbits && !pos) atomicMax(&s_cet, __float_as_uint(ce));
        }
    }
    float sp = blk_reduce_f(sum_pos, red_f, tid);
    float sg = blk_reduce_f(sum_gt, red_f, tid);
    int cg = blk_reduce_i(cgt, red_i, tid);
    if (tid == 0) {
        float tie = 0.0f;
        // floss is strictly monotone in ce, so all non-pos ties at t share one ce.
        if (k > 0 && tbits != 0u) tie = (float)(k - cg) * __uint_as_float(s_cet);
        cls_out[row] = sp + sg + tie;
        np_out[row] = np;
    }
}

// ---------- K3: final deterministic reduction + normalize ----------
__global__ __launch_bounds__(NTH) void k3_final(
    const float* __restrict__ loc_part, const float* __restrict__ cls_part,
    const int* __restrict__ np_part, float* __restrict__ out) {
    __shared__ float red_f[NTH];
    __shared__ int red_i[NTH];
    int tid = threadIdx.x;
    float ls = 0.0f;
    for (int i = tid; i < NBLK1; i += NTH) ls += loc_part[i];
    float loc = blk_reduce_f(ls, red_f, tid);
    float cs = 0.0f;
    for (int i = tid; i < B_; i += NTH) cs += cls_part[i];
    float cls = blk_reduce_f(cs, red_f, tid);
    int nl = 0;
    for (int i = tid; i < B_; i += NTH) nl += np_part[i];
    int N = blk_reduce_i(nl, red_i, tid);
    if (tid == 0) {
        float n = (float)N;
        out[0] = loc / n;
        out[1] = cls / n;
    }
}

extern "C" void kernel_launch(void* const* d_in, const int* in_sizes, int n_in,
                              void* d_out, int out_size, void* d_ws, size_t ws_size,
                              hipStream_t stream) {
    const float* pred_loc   = (const float*)d_in[0];
    const float* pred_label = (const float*)d_in[1];
    const float* gt_loc     = (const float*)d_in[2];
    const int*   gt_label   = (const int*)d_in[3];
    float* out = (float*)d_out;

    float* ce       = (float*)d_ws;                 // B*A floats (~4.47 MB)
    float* loc_part = ce + (size_t)B_ * A_;         // NBLK1 floats
    float* cls_part = loc_part + NBLK1;             // B floats
    int*   np_part  = (int*)(cls_part + B_);        // B ints

    k1_ce_loc<<<NBLK1, NTH, 0, stream>>>(pred_loc, pred_label, gt_loc, gt_label, ce, loc_part);
    k2_mine<<<B_, NTH, 0, stream>>>(ce, gt_label, cls_part, np_part);
    k3_final<<<1, NTH, 0, stream>>>(loc_part, cls_part, np_part, out);
}